// FirstSpikeDetector_23407571764162
// MI455X (gfx1250) — compile-verified
//
#include <hip/hip_runtime.h>
#include <stdint.h>

// First-spike detector: out[b][t] = spike[b][t] AND no spike at t' < t in row b.
// Memory-bound streaming problem (256 MiB @ 23.3 TB/s ~= 11.5 us; ~6 us with
// early-exit reads). One wave32 per row; ballot-based prefix-OR replaces the
// T-serial scan with 16 wave-uniform steps. Reads go through the CDNA5 async
// global->LDS data mover (ASYNCcnt), double-buffered per wave. Writes are
// non-temporal b128 (output > L2, never re-read).

typedef float v4f __attribute__((ext_vector_type(4)));

#define WPB   8                 // waves per block (256 threads)
#define LANES 32                // wave32
#define PIPE  2                 // async double buffer depth
#define CHUNK 128               // floats per wave-chunk: 32 lanes * 4 floats (b128/lane)

__global__ __launch_bounds__(WPB * LANES)
void first_spike_kernel(const float* __restrict__ in,
                        float* __restrict__ out,
                        int T, int rows) {
    // Per-wave private double buffers: WPB * PIPE * 512B = 8 KB per block.
    __shared__ v4f smem[WPB * PIPE * LANES];

    const int wave = threadIdx.x >> 5;
    const int lane = threadIdx.x & 31;
    const int row  = blockIdx.x * WPB + wave;
    if (row >= rows) return;                 // wave-uniform

    const float* rowp = in  + (size_t)row * (size_t)T;
    float*       orow = out + (size_t)row * (size_t)T;
    const int nchunk  = T / CHUNK;           // 2048/128 = 16

    // Async copy one 512B chunk (b128 per lane) from global to this wave's
    // LDS buffer. Tracked by ASYNCcnt; no workgroup barrier needed since each
    // wave owns its buffers and waits on its own counter.
    auto issue = [&](int c, int buf) {
        const v4f* g = (const v4f*)(rowp + (size_t)c * CHUNK) + lane;
        unsigned ldsaddr =
            (unsigned)(uintptr_t)&smem[(wave * PIPE + buf) * LANES + lane];
        asm volatile("global_load_async_to_lds_b128 %0, %1, off"
                     :: "v"(ldsaddr), "v"(g) : "memory");
    };

    unsigned seen = 0;          // wave-uniform: OR of all ballots so far
    issue(0, 0);

    for (int c = 0; c < nchunk; ++c) {
        v4f x = {0.0f, 0.0f, 0.0f, 0.0f};
        if (seen == 0) {        // uniform branch: after first spike, stop reading
            if (c + 1 < nchunk) {
                issue(c + 1, (c + 1) & (PIPE - 1));
                asm volatile("s_wait_asynccnt 1" ::: "memory");
            } else {
                asm volatile("s_wait_asynccnt 0" ::: "memory");
            }
            x = smem[(wave * PIPE + (c & (PIPE - 1))) * LANES + lane];
        }

        // Per-lane spike bits (inputs are exact 0.0/1.0)
        unsigned b0 = x.x != 0.0f;
        unsigned b1 = x.y != 0.0f;
        unsigned b2 = x.z != 0.0f;
        unsigned b3 = x.w != 0.0f;
        unsigned any = b0 | b1 | b2 | b3;

#if __has_builtin(__builtin_amdgcn_ballot_w32)
        unsigned m = __builtin_amdgcn_ballot_w32(any != 0);
#else
        unsigned m = (unsigned)__ballot(any != 0);
#endif
        unsigned lower = m & ((1u << lane) - 1u);
        bool elig = (seen == 0) && (lower == 0);   // no spike anywhere before this lane

        v4f o;
        o.x = (elig && b0)                        ? 1.0f : 0.0f;
        o.y = (elig && !b0 && b1)                 ? 1.0f : 0.0f;
        o.z = (elig && !(b0 | b1) && b2)          ? 1.0f : 0.0f;
        o.w = (elig && !(b0 | b1 | b2) && b3)     ? 1.0f : 0.0f;

        __builtin_nontemporal_store(o, (v4f*)(orow + (size_t)c * CHUNK) + lane);

        seen |= m;              // stays wave-uniform (ballot result is scalar)
    }
}

extern "C" void kernel_launch(void* const* d_in, const int* in_sizes, int n_in,
                              void* d_out, int out_size, void* d_ws, size_t ws_size,
                              hipStream_t stream) {
    (void)n_in; (void)out_size; (void)d_ws; (void)ws_size;
    const float* in  = (const float*)d_in[0];
    float*       out = (float*)d_out;

    const int T    = 2048;                   // time axis length (reference)
    const int rows = in_sizes[0] / T;        // 16384

    dim3 block(WPB * LANES);                 // 256 threads = 8 wave32
    dim3 grid((rows + WPB - 1) / WPB);       // one wave per row
    hipLaunchKernelGGL(first_spike_kernel, grid, block, 0, stream,
                       in, out, T, rows);
}